// OptNetIneq_13924283974423
// MI455X (gfx1250) — compile-verified
//
#include <hip/hip_runtime.h>

// ---------------------------------------------------------------------------
// Batched QP interior-point solver (OptNet-style) for MI455X / gfx1250.
//
// Per batch item b (NITER=10 fixed iterations):
//   K = diag(QPEN + d1) + G2^T D2 G2   (729x729)   -> Woodbury:
//   K^{-1} r = A^{-1} r - A^{-1} G2^T S^{-1} G2 A^{-1} r
//   S = diag(1/d2) + G2 diag(1/a) G2^T             (128x128 SPD)
//
// CDNA5 feature usage:
//  * V_WMMA_F32_16X16X4_F32 for the 128x128x729 Gram product (8x8 tile grid,
//    one tile row per wave32, K staged through LDS in 64-wide chunks).
//  * Tensor Data Mover (TENSOR_LOAD_TO_LDS) to DMA each full G2 K-chunk into
//    LDS with hardware row padding (64 dwords data + 2 dwords pad = pitch 66),
//    synchronized with s_wait_tensorcnt.
//  * global_load_async_to_lds_b32 (ASYNCcnt) for the partial tail chunk.
// One workgroup (256 threads = 8 wave32) per batch item; all state in LDS.
// ---------------------------------------------------------------------------

typedef __attribute__((ext_vector_type(2))) float v2f;
typedef __attribute__((ext_vector_type(8))) float v8f;
typedef __attribute__((ext_vector_type(4))) unsigned int u32x4;
typedef __attribute__((ext_vector_type(4))) int i32x4;
typedef __attribute__((ext_vector_type(8))) int i32x8;

#define NXQ 729          // decision vars
#define M2Q 128          // dense inequality rows
#define MT  857          // total constraints (729 + 128)
#define NXP 736          // padded NX (even, covers k-step overread)
#define MP  864          // padded MT
#define SP  130          // S row pitch (floats): avoids 64-bank conflicts
#define KC  64           // K-chunk staged in LDS per Gram pass
#define KP  66           // G2 chunk row pitch (floats), even for float2 loads
#define NITER 10
#define QPEN 0.1f
#define SIGMA 0.1f
#define NTHREADS 256

#if defined(__HIP_DEVICE_COMPILE__) && __has_builtin(__builtin_amdgcn_tensor_load_to_lds)
#define HAVE_TDM 1
#else
#define HAVE_TDM 0
#endif

struct Smem {
  float S[M2Q * SP];     // 128x128 SPD matrix / Cholesky factor (pitch SP)
  float G2c[M2Q * KP];   // staged G2 K-chunk (zero padded)
  float z[NXP], p[NXP], rhs[NXP], t[NXP], dz[NXP], ainv[NXP];
  float sv[MP], lam[MP], dv[MP], rp[MP], vv[MP], dsv[MP], dlv[MP];
  float gz2[M2Q], y[M2Q], h2[M2Q], gd2[M2Q];
  float red[NTHREADS];
};

__global__ __launch_bounds__(NTHREADS)
void optnet_ipm_kernel(const float* __restrict__ puzzles,
                       const float* __restrict__ G2,
                       const float* __restrict__ z2,
                       const float* __restrict__ s2,
                       float* __restrict__ out)
{
  extern __shared__ char smem_raw[];
  Smem& sm = *reinterpret_cast<Smem*>(smem_raw);

  const int tid  = threadIdx.x;
  const int b    = blockIdx.x;
  const int lane = tid & 31;
  const int wave = tid >> 5;        // 0..7 -> tile row of S
  const int half = lane >> 4;       // K-half selector inside a wmma fragment
  const int nr   = lane & 15;       // M/N index inside a 16-wide tile

  // ---- init state: z = 0, s = 1, lam = 1, p = -puzzle, h2 = G2 z2 + s2 ----
  for (int i = tid; i < NXP; i += NTHREADS) {
    sm.z[i]    = 0.f;
    sm.p[i]    = (i < NXQ) ? -puzzles[b * NXQ + i] : 0.f;
    sm.ainv[i] = 0.f;                      // zero pad keeps wmma tail clean
  }
  for (int i = tid; i < MP; i += NTHREADS) { sm.sv[i] = 1.f; sm.lam[i] = 1.f; }
  if (tid < M2Q) {
    const float* row = G2 + tid * NXQ;
    float acc = 0.f;
    for (int i = 0; i < NXQ; ++i) acc += row[i] * z2[i];
    sm.h2[tid] = acc + s2[tid];
  }
  __syncthreads();

  for (int it = 0; it < NITER; ++it) {
    // ---- gz2 = G2 z ------------------------------------------------------
    if (tid < M2Q) {
      const float* row = G2 + tid * NXQ;
      float acc = 0.f;
      for (int i = 0; i < NXQ; ++i) acc += row[i] * sm.z[i];
      sm.gz2[tid] = acc;
    }
    __syncthreads();

    // ---- primal residual rp = Gz + s - h  (h1 = 0) -----------------------
    for (int i = tid; i < NXQ; i += NTHREADS) sm.rp[i] = sm.sv[i] - sm.z[i];
    for (int j = tid; j < M2Q; j += NTHREADS)
      sm.rp[NXQ + j] = sm.gz2[j] + sm.sv[NXQ + j] - sm.h2[j];

    // ---- mu = mean(s .* lam) --------------------------------------------
    float part = 0.f;
    for (int i = tid; i < MT; i += NTHREADS) part += sm.sv[i] * sm.lam[i];
    sm.red[tid] = part;
    __syncthreads();
    for (int off = NTHREADS / 2; off > 0; off >>= 1) {
      if (tid < off) sm.red[tid] += sm.red[tid + off];
      __syncthreads();
    }
    const float mu = sm.red[0] / (float)MT;
    __syncthreads();

    // ---- d = lam/s ;  v = SIGMA*mu/s - lam + d .* rp ---------------------
    for (int i = tid; i < MT; i += NTHREADS) {
      const float d = sm.lam[i] / sm.sv[i];
      sm.dv[i] = d;
      sm.vv[i] = SIGMA * mu / sm.sv[i] - sm.lam[i] + d * sm.rp[i];
    }
    __syncthreads();

    // ---- rhs = -(QPEN z + p - lam1 - v1 + G2^T (lam2 + v2)), ainv --------
    for (int i = tid; i < NXQ; i += NTHREADS) {
      float acc = 0.f;
      for (int j = 0; j < M2Q; ++j)
        acc += G2[j * NXQ + i] * (sm.lam[NXQ + j] + sm.vv[NXQ + j]);
      const float rd = QPEN * sm.z[i] + sm.p[i] - sm.lam[i] - sm.vv[i] + acc;
      sm.rhs[i]  = -rd;
      sm.ainv[i] = 1.f / (QPEN + sm.dv[i]);
    }
    __syncthreads();

    // ---- Gram: S = (G2 diag(ainv)) G2^T  via V_WMMA_F32_16X16X4_F32 ------
    v8f acc8[8];
#pragma unroll
    for (int c = 0; c < 8; ++c)
#pragma unroll
      for (int j = 0; j < 8; ++j) acc8[c][j] = 0.f;

    const int arow = (wave << 4) + nr;          // S row handled by this lane
    for (int k0 = 0; k0 < NXQ; k0 += KC) {
      __syncthreads();
      const bool fullChunk = (NXQ - k0) >= KC;
#if HAVE_TDM
      if (fullChunk) {
        // ---- Tensor Data Mover: DMA 128x64 f32 tile (row stride 729) -----
        // into LDS with pad_amount=2 dwords every pad_interval=64 dwords
        // (-> LDS row pitch KP=66 floats).  Issued by wave 0 only.
        if (wave == 0) {
          const unsigned long long ga =
              (unsigned long long)(uintptr_t)G2 + (unsigned long long)k0 * 4ull;
          u32x4 g0;
          g0[0] = 1u;                                      // count=1, no gather
          g0[1] = (unsigned int)(uintptr_t)&sm.G2c[0];     // lds_addr
          g0[2] = (unsigned int)ga;                        // global_addr[31:0]
          g0[3] = (unsigned int)((ga >> 32) & 0x01FFFFFFull) | 0x80000000u; // type=2
          i32x8 g1;
          g1[0] = (int)((2u << 16)        // data_size = 4B
                      | (1u << 20)        // pad_enable
                      | (5u << 22)        // pad_interval: 64 dwords
                      | (1u << 25));      // pad_amount: 2 dwords
          g1[1] = (int)((unsigned)NXQ << 16);   // tensor_dim0[15:0]
          g1[2] = (int)((unsigned)M2Q << 16);   // tensor_dim1[15:0]
          g1[3] = (int)((unsigned)KC << 16);    // tile_dim0
          g1[4] = (int)M2Q;                     // tile_dim1
          g1[5] = (int)NXQ;                     // tensor_dim0_stride[31:0]
          g1[6] = 0;
          g1[7] = 0;
          const i32x4 gz_ = {0, 0, 0, 0};       // groups 2/3: dims 2..4 unused
#if defined(__clang_major__) && (__clang_major__ >= 23)
          const i32x8 gz8 = {0, 0, 0, 0, 0, 0, 0, 0};
          __builtin_amdgcn_tensor_load_to_lds(g0, g1, gz_, gz_, gz8, 0);
#else
          __builtin_amdgcn_tensor_load_to_lds(g0, g1, gz_, gz_, 0);
#endif
          __builtin_amdgcn_s_wait_tensorcnt(0);
        }
      } else
#endif
      {
        // ---- fallback / tail chunk: async global->LDS copy, zero padded --
        for (int idx = tid; idx < M2Q * KC; idx += NTHREADS) {
          const int r = idx >> 6;                  // /KC (64)
          const int k = idx & (KC - 1);
          const int gk = k0 + k;
          if (gk < NXQ) {
            const unsigned int ldsoff =
                (unsigned int)(uintptr_t)&sm.G2c[r * KP + k];
            const unsigned long long gaddr =
                (unsigned long long)(uintptr_t)&G2[r * NXQ + gk];
            asm volatile("global_load_async_to_lds_b32 %0, %1, off"
                         :: "v"(ldsoff), "v"(gaddr) : "memory");
          } else {
            sm.G2c[r * KP + k] = 0.f;
          }
        }
        asm volatile("s_wait_asynccnt 0x0" ::: "memory");
      }
      // prefetch next chunk of this lane's row into cache
      if (k0 + KC < NXQ)
        __builtin_prefetch(&G2[arow * NXQ + k0 + KC], 0, 0);
      __syncthreads();

      int kcEff = NXQ - k0; if (kcEff > KC) kcEff = KC;
      const int kend = (kcEff + 3) & ~3;         // round up to K=4 steps
      for (int kk = 0; kk < kend; kk += 4) {
        const int kk2 = kk + (half << 1);
        // A fragment: W[m,k] = G2[m,k] * ainv[k]
        const float2 ga = *(const float2*)&sm.G2c[arow * KP + kk2];
        const float2 ai = *(const float2*)&sm.ainv[k0 + kk2];
        v2f a; a.x = ga.x * ai.x; a.y = ga.y * ai.y;
#pragma unroll
        for (int c = 0; c < 8; ++c) {
          // B fragment: B[k,n] = G2[16c+n, k]
          const float2 gb = *(const float2*)&sm.G2c[((c << 4) + nr) * KP + kk2];
          v2f bb; bb.x = gb.x; bb.y = gb.y;
          acc8[c] = __builtin_amdgcn_wmma_f32_16x16x4_f32(
              false, a, false, bb, (short)0, acc8[c], false, false);
        }
      }
    }
    __syncthreads();
    // store accumulators: D layout: VGPR j -> M = j + 8*half, N = nr
#pragma unroll
    for (int c = 0; c < 8; ++c) {
#pragma unroll
      for (int j = 0; j < 8; ++j) {
        const int m = (wave << 4) + j + (half << 3);
        sm.S[m * SP + (c << 4) + nr] = acc8[c][j];
      }
    }
    __syncthreads();
    if (tid < M2Q)   // S += diag(1/d2)
      sm.S[tid * SP + tid] += sm.sv[NXQ + tid] / sm.lam[NXQ + tid];
    __syncthreads();

    // ---- in-LDS Cholesky of S (symmetric trailing update, row/thread) ----
    for (int k = 0; k < M2Q; ++k) {
      if (tid == 0) sm.S[k * SP + k] = sqrtf(sm.S[k * SP + k]);
      __syncthreads();
      const float dkk = sm.S[k * SP + k];
      for (int i = k + 1 + tid; i < M2Q; i += NTHREADS) sm.S[i * SP + k] /= dkk;
      __syncthreads();
      const int i = k + 1 + tid;
      if (i < M2Q) {
        const float lik = sm.S[i * SP + k];
        for (int j = k + 1; j < M2Q; ++j)
          sm.S[i * SP + j] -= lik * sm.S[j * SP + k];
      }
      __syncthreads();
    }

    // ---- t = ainv .* rhs ;  w = G2 t  ------------------------------------
    for (int i = tid; i < NXQ; i += NTHREADS) sm.t[i] = sm.ainv[i] * sm.rhs[i];
    __syncthreads();
    if (tid < M2Q) {
      const float* row = G2 + tid * NXQ;
      float acc = 0.f;
      for (int i = 0; i < NXQ; ++i) acc += row[i] * sm.t[i];
      sm.y[tid] = acc;       // y <- w, solved in place
    }
    __syncthreads();

    // ---- solve S y = w:  L (fwd) then L^T (bwd) --------------------------
    for (int k = 0; k < M2Q; ++k) {
      if (tid == 0) sm.y[k] /= sm.S[k * SP + k];
      __syncthreads();
      const float yk = sm.y[k];
      for (int i = k + 1 + tid; i < M2Q; i += NTHREADS)
        sm.y[i] -= sm.S[i * SP + k] * yk;
      __syncthreads();
    }
    for (int k = M2Q - 1; k >= 0; --k) {
      if (tid == 0) sm.y[k] /= sm.S[k * SP + k];
      __syncthreads();
      const float yk = sm.y[k];
      for (int i = tid; i < k; i += NTHREADS)
        sm.y[i] -= sm.S[k * SP + i] * yk;
      __syncthreads();
    }

    // ---- dz = t - ainv .* (G2^T y) ---------------------------------------
    for (int i = tid; i < NXQ; i += NTHREADS) {
      float acc = 0.f;
      for (int j = 0; j < M2Q; ++j) acc += G2[j * NXQ + i] * sm.y[j];
      sm.dz[i] = sm.t[i] - sm.ainv[i] * acc;
    }
    __syncthreads();

    // ---- gd2 = G2 dz ;  ds, dlam -----------------------------------------
    if (tid < M2Q) {
      const float* row = G2 + tid * NXQ;
      float acc = 0.f;
      for (int i = 0; i < NXQ; ++i) acc += row[i] * sm.dz[i];
      sm.gd2[tid] = acc;
    }
    __syncthreads();
    for (int i = tid; i < NXQ; i += NTHREADS) {
      sm.dsv[i] = sm.dz[i] - sm.rp[i];              // ds1 = -rp1 + dz
      sm.dlv[i] = sm.vv[i] - sm.dv[i] * sm.dz[i];   // dlam1 = v1 - d1*dz
    }
    for (int j = tid; j < M2Q; j += NTHREADS) {
      sm.dsv[NXQ + j] = -sm.rp[NXQ + j] - sm.gd2[j];
      sm.dlv[NXQ + j] = sm.vv[NXQ + j] + sm.dv[NXQ + j] * sm.gd2[j];
    }
    __syncthreads();

    // ---- alpha = min(1, 0.99 * fraction-to-boundary) ---------------------
    float rmin = 3.4e38f;
    for (int i = tid; i < MT; i += NTHREADS) {
      const float dsi = sm.dsv[i];
      if (dsi < 0.f) rmin = fminf(rmin, -sm.sv[i] / dsi);
      const float dli = sm.dlv[i];
      if (dli < 0.f) rmin = fminf(rmin, -sm.lam[i] / dli);
    }
    sm.red[tid] = rmin;
    __syncthreads();
    for (int off = NTHREADS / 2; off > 0; off >>= 1) {
      if (tid < off) sm.red[tid] = fminf(sm.red[tid], sm.red[tid + off]);
      __syncthreads();
    }
    const float alpha = fminf(1.f, 0.99f * sm.red[0]);
    __syncthreads();

    // ---- state update ----------------------------------------------------
    for (int i = tid; i < NXQ; i += NTHREADS) sm.z[i] += alpha * sm.dz[i];
    for (int i = tid; i < MT; i += NTHREADS) {
      sm.sv[i]  += alpha * sm.dsv[i];
      sm.lam[i] += alpha * sm.dlv[i];
    }
    __syncthreads();
  }

  // ---- write result ------------------------------------------------------
  for (int i = tid; i < NXQ; i += NTHREADS) out[b * NXQ + i] = sm.z[i];
}

extern "C" void kernel_launch(void* const* d_in, const int* in_sizes, int n_in,
                              void* d_out, int out_size, void* d_ws, size_t ws_size,
                              hipStream_t stream) {
  const float* puzzles = (const float*)d_in[0];   // (B,9,9,9) f32
  const float* G2      = (const float*)d_in[1];   // (128,729) f32
  const float* z2      = (const float*)d_in[2];   // (729,)    f32
  const float* s2      = (const float*)d_in[3];   // (128,)    f32
  float* out = (float*)d_out;

  const int Bn = in_sizes[0] / NXQ;               // 16 puzzles
  const size_t shmem = sizeof(Smem);              // ~146 KB dynamic LDS
  (void)hipFuncSetAttribute((const void*)optnet_ipm_kernel,
                            hipFuncAttributeMaxDynamicSharedMemorySize,
                            (int)shmem);
  optnet_ipm_kernel<<<dim3(Bn), dim3(NTHREADS), shmem, stream>>>(
      puzzles, G2, z2, s2, out);
}